// UpFirDn2d_35974646071930
// MI455X (gfx1250) — compile-verified
//
#include <hip/hip_runtime.h>

#define AS1 __attribute__((address_space(1)))
#define AS3 __attribute__((address_space(3)))

typedef float float4v __attribute__((ext_vector_type(4)));
typedef int   v4i     __attribute__((ext_vector_type(4)));

// upfirdn2d, up=2, down=1, pad=(2,1), 4x4 kernel.
// Per output row y (m = y>>1):
//   y even: taps k={0,2} on input rows {m-1, m}
//   y odd : taps k={1,3} on input rows {m,   m+1}
// Same per column. Weights use the flipped kernel w[k][l] = K[3-k][3-l].
__global__ __launch_bounds__(256) void upfirdn2d_up2_kernel(
    const float* __restrict__ x,    // [planes, 64, 64]
    const float* __restrict__ k4,   // [4, 4]
    float* __restrict__ out)        // [planes, 128, 128]
{
  __shared__ float tile[64 * 64];   // one full input plane, 16 KB
  __shared__ float kw[16];

  const int t     = threadIdx.x;
  const int plane = blockIdx.x;
  const float* gin  = x   + (size_t)plane * (64 * 64);
  float*       gout = out + (size_t)plane * (128 * 128);

  // ---- async stage: global -> LDS, 16 B per lane per instruction ----
  // 1024 float4s total; 256 threads x 4 issues.
#pragma unroll
  for (int i = 0; i < 4; ++i) {
    const int v4 = t + i * 256;                       // float4 index
    AS1 v4i* gp = (AS1 v4i*)(gin + v4 * 4);
    AS3 v4i* lp = (AS3 v4i*)(&tile[v4 * 4]);
#if defined(__has_builtin) && __has_builtin(__builtin_amdgcn_global_load_async_to_lds_b128)
    __builtin_amdgcn_global_load_async_to_lds_b128(gp, lp, 0, 0);
#else
    const unsigned           l32 = (unsigned)(unsigned long long)lp;
    const unsigned long long g64 = (unsigned long long)gp;
    asm volatile("global_load_async_to_lds_b128 %0, %1, off"
                 :: "v"(l32), "v"(g64) : "memory");
#endif
  }
  if (t < 16) kw[t] = k4[t];

#if defined(__has_builtin) && __has_builtin(__builtin_amdgcn_s_wait_asynccnt)
  __builtin_amdgcn_s_wait_asynccnt(0);
#else
  asm volatile("s_wait_asynccnt 0" ::: "memory");
#endif
  __syncthreads();

  // ---- compute: each wave owns rows y = 8p + tg, lane tx owns out cols 4tx..4tx+3
  const int tx = t & 31;       // float4 column index (0..31)
  const int tg = t >> 5;       // wave id (0..7)
  const int po = tg & 1;       // row parity: y = 8p + tg -> parity == tg&1, constant

  const int kkA = po ? 1 : 0;  // first vertical tap
  const int kkB = kkA + 2;     // second vertical tap
  const int ra  = (3 - kkA) * 4;
  const int rb  = (3 - kkB) * 4;
  // w(k,l) = kw[(3-k)*4 + (3-l)]
  const float wA0 = kw[ra + 3], wA1 = kw[ra + 2], wA2 = kw[ra + 1], wA3 = kw[ra + 0];
  const float wB0 = kw[rb + 3], wB1 = kw[rb + 2], wB2 = kw[rb + 1], wB3 = kw[rb + 0];

  const int c0 = 2 * tx - 1;   // leftmost needed input column

#pragma unroll 4
  for (int p = 0; p < 16; ++p) {
    const int y  = p * 8 + tg;
    const int m  = y >> 1;
    const int rA = po ? m     : m - 1;   // in [-1, 63]
    const int rB = po ? m + 1 : m;       // in [0, 64]
    const bool vA = (rA >= 0);
    const bool vB = (rB <= 63);

    const float* rowA = &tile[(vA ? rA : 0) * 64];
    const float* rowB = &tile[(vB ? rB : 0) * 64];

    float a0, a1, a2, a3, b0, b1, b2, b3;
    if (vA) {
      a1 = rowA[c0 + 1];
      a2 = rowA[c0 + 2];
      a0 = (tx == 0)  ? 0.0f : rowA[c0];
      a3 = (tx == 31) ? 0.0f : rowA[c0 + 3];
    } else {
      a0 = a1 = a2 = a3 = 0.0f;
    }
    if (vB) {
      b1 = rowB[c0 + 1];
      b2 = rowB[c0 + 2];
      b0 = (tx == 0)  ? 0.0f : rowB[c0];
      b3 = (tx == 31) ? 0.0f : rowB[c0 + 3];
    } else {
      b0 = b1 = b2 = b3 = 0.0f;
    }

    float4v o;
    // x = 4tx+0: even phase, n = 2tx   -> cols {n-1, n}   = {c0,   c0+1}
    o.x = wA0 * a0 + wA2 * a1 + wB0 * b0 + wB2 * b1;
    // x = 4tx+1: odd phase,  n = 2tx   -> cols {n, n+1}   = {c0+1, c0+2}
    o.y = wA1 * a1 + wA3 * a2 + wB1 * b1 + wB3 * b2;
    // x = 4tx+2: even phase, n = 2tx+1 -> cols {n-1, n}   = {c0+1, c0+2}
    o.z = wA0 * a1 + wA2 * a2 + wB0 * b1 + wB2 * b2;
    // x = 4tx+3: odd phase,  n = 2tx+1 -> cols {n, n+1}   = {c0+2, c0+3}
    o.w = wA1 * a2 + wA3 * a3 + wB1 * b2 + wB3 * b3;

    // streamed output (268 MB > L2): non-temporal 128-bit store, coalesced
    __builtin_nontemporal_store(o, (float4v*)(gout + y * 128 + tx * 4));
  }
}

extern "C" void kernel_launch(void* const* d_in, const int* in_sizes, int n_in,
                              void* d_out, int out_size, void* d_ws, size_t ws_size,
                              hipStream_t stream) {
  (void)n_in; (void)out_size; (void)d_ws; (void)ws_size;
  const float* x = (const float*)d_in[0];
  const float* k = (const float*)d_in[1];
  float* out = (float*)d_out;
  const int planes = in_sizes[0] / (64 * 64);  // 8*512 = 4096
  upfirdn2d_up2_kernel<<<dim3(planes), dim3(256), 0, stream>>>(x, k, out);
}